// MultiBoxLoss_combined_24481313587723
// MI455X (gfx1250) — compile-verified
//
#include <hip/hip_runtime.h>
#include <hip/hip_bf16.h>

#define Bn 64
#define Pn 16320
#define Cn 81
#define On 50

typedef __attribute__((ext_vector_type(2))) float v2f;
typedef __attribute__((ext_vector_type(8))) float v8f;

__device__ __forceinline__ float sl1f(float d) {
    float a = fabsf(d);
    return (a < 1.0f) ? 0.5f * d * d : a - 0.5f;
}

// Async copy one dword from global memory into LDS (CDNA5 ASYNCcnt path).
// GV addressing: LDS address from the VDST VGPR, global address from a 64-bit VADDR pair.
__device__ __forceinline__ void async_g2l_b32(float* lds_generic, const float* gaddr) {
    // Low 32 bits of the flat shared-aperture address are the LDS byte offset.
    unsigned lds_off = (unsigned)(unsigned long long)lds_generic;
    asm volatile("global_load_async_to_lds_b32 %0, %1, off"
                 :: "v"(lds_off), "v"(gaddr) : "memory");
}

__device__ __forceinline__ void wait_asynccnt0() {
#if __has_builtin(__builtin_amdgcn_s_wait_asynccnt)
    __builtin_amdgcn_s_wait_asynccnt(0);
#else
    asm volatile("s_wait_asynccnt 0x0" ::: "memory");
#endif
}

// ---------------------------------------------------------------------------
// K0: zero accumulators
// ---------------------------------------------------------------------------
__global__ void k_zero(float* accum_f, int* accum_i, int* num_pos) {
    int t = threadIdx.x;
    if (t < 8) accum_f[t] = 0.0f;
    if (t < 4) accum_i[t] = 0;
    if (t < Bn) num_pos[t] = 0;
}

// ---------------------------------------------------------------------------
// K1: SSD matching per batch image (one workgroup per image).
//  - target table staged into LDS with global_load_async_to_lds_b32
//  - per-prior best truth (first-max over O) kept in tmp arrays
//  - per-truth best prior via packed 64-bit LDS atomicMax (first-max tie-break)
//  - forced matches applied from an LDS list (ascending scan => last wins)
//  - emits conf_t, smooth-L1 loc loss over positives, positive counts
// ---------------------------------------------------------------------------
__global__ __launch_bounds__(1024) void k_match(
    const float* __restrict__ loc, const float* __restrict__ priors,
    const float* __restrict__ targets, int* __restrict__ conf_t,
    float* __restrict__ tmp_ov, int* __restrict__ tmp_ti,
    int* __restrict__ num_pos, float* __restrict__ accum_f,
    int* __restrict__ accum_i) {
    const int b = blockIdx.x;
    const int tid = threadIdx.x;

    __shared__ float s_raw[On * 5 + 6];
    __shared__ float s_x1[On], s_y1[On], s_x2[On], s_y2[On], s_area[On];
    __shared__ int s_label[On];
    __shared__ unsigned long long s_best[On];
    __shared__ int s_forced[On];
    __shared__ float rf[1024];
    __shared__ int ri[1024];

    // Stage the 50x5 target table into LDS asynchronously (ASYNCcnt path).
    if (tid < On * 5)
        async_g2l_b32(&s_raw[tid], targets + (size_t)b * On * 5 + tid);
    wait_asynccnt0();
    __syncthreads();

    if (tid < On) {
        float x1 = s_raw[tid * 5 + 0], y1 = s_raw[tid * 5 + 1];
        float x2 = s_raw[tid * 5 + 2], y2 = s_raw[tid * 5 + 3];
        s_x1[tid] = x1; s_y1[tid] = y1; s_x2[tid] = x2; s_y2[tid] = y2;
        s_area[tid] = (x2 - x1) * (y2 - y1);
        s_label[tid] = (int)s_raw[tid * 5 + 4];
        s_best[tid] = 0ull;
    }
    __syncthreads();

    // Phase B: per-prior best truth + per-truth best prior
    for (int p = tid; p < Pn; p += blockDim.x) {
        float cx = priors[p * 4 + 0], cy = priors[p * 4 + 1];
        float pw = priors[p * 4 + 2], ph = priors[p * 4 + 3];
        float px1 = cx - 0.5f * pw, py1 = cy - 0.5f * ph;
        float px2 = cx + 0.5f * pw, py2 = cy + 0.5f * ph;
        float area_p = pw * ph;
        float bov = -1.0f; int bti = 0;
        for (int o = 0; o < On; ++o) {
            float lx = fmaxf(s_x1[o], px1), ly = fmaxf(s_y1[o], py1);
            float rx = fminf(s_x2[o], px2), ry = fminf(s_y2[o], py2);
            float iw = fmaxf(rx - lx, 0.0f), ih = fmaxf(ry - ly, 0.0f);
            float inter = iw * ih;
            float ov = inter / (s_area[o] + area_p - inter);
            if (ov > bov) { bov = ov; bti = o; }
            unsigned long long key =
                ((unsigned long long)__float_as_uint(ov) << 32) |
                (unsigned long long)(0xFFFFFFFFu - (unsigned)p);
            if (key > s_best[o]) atomicMax(&s_best[o], key);
        }
        size_t idx = (size_t)b * Pn + p;
        tmp_ov[idx] = bov;
        tmp_ti[idx] = bti;
    }
    __syncthreads();

    if (tid == 0) {
        for (int o = 0; o < On; ++o)
            s_forced[o] = (int)(0xFFFFFFFFu - (unsigned)(s_best[o] & 0xFFFFFFFFull));
    }
    __syncthreads();

    // Phase D: forced overrides, conf_t, localization loss (same thread reads
    // back its own tmp stores -> same-wave store->load ordering guaranteed)
    float lsum = 0.0f; int pcnt = 0;
    for (int p = tid; p < Pn; p += blockDim.x) {
        size_t idx = (size_t)b * Pn + p;
        float ov = tmp_ov[idx];
        int ti = tmp_ti[idx];
        for (int o = 0; o < On; ++o)
            if (s_forced[o] == p) { ov = 2.0f; ti = o; }
        int conf = (ov < 0.5f) ? 0 : (s_label[ti] + 1);
        conf_t[idx] = conf;
        if (conf > 0) {
            float mx1 = s_x1[ti], my1 = s_y1[ti], mx2 = s_x2[ti], my2 = s_y2[ti];
            float cx = priors[p * 4 + 0], cy = priors[p * 4 + 1];
            float pw = priors[p * 4 + 2], ph = priors[p * 4 + 3];
            float gx = ((mx1 + mx2) * 0.5f - cx) / (0.1f * pw);
            float gy = ((my1 + my2) * 0.5f - cy) / (0.1f * ph);
            float gw = logf((mx2 - mx1) / pw) / 0.2f;
            float gh = logf((my2 - my1) / ph) / 0.2f;
            const float* l = loc + idx * 4;
            lsum += sl1f(l[0] - gx) + sl1f(l[1] - gy) +
                    sl1f(l[2] - gw) + sl1f(l[3] - gh);
            pcnt++;
        }
    }

    rf[tid] = lsum; ri[tid] = pcnt;
    __syncthreads();
    for (int s = blockDim.x / 2; s > 0; s >>= 1) {
        if (tid < s) { rf[tid] += rf[tid + s]; ri[tid] += ri[tid + s]; }
        __syncthreads();
    }
    if (tid == 0) {
        atomicAdd(&accum_f[0], rf[0]);
        atomicAdd(&num_pos[b], ri[0]);
        atomicAdd(&accum_i[0], ri[0]);
    }
}

// ---------------------------------------------------------------------------
// K2: streaming CE pass (HBM-bound: 338 MB of conf_data in one coalesced
// pass => ~15us floor at 23.3 TB/s). One wave32 per anchor row; lanes cover
// the 81 logits; butterfly shuffles give logsumexp. global_prefetch_b8 keeps
// the next row in flight.
// ---------------------------------------------------------------------------
__global__ __launch_bounds__(256) void k_ce(
    const float* __restrict__ conf, const float* __restrict__ obj,
    const int* __restrict__ conf_t, float* __restrict__ masked_c,
    float* __restrict__ masked_o, float* __restrict__ accum_f) {
    const int lane = threadIdx.x & 31;
    const int wave = blockIdx.x * (blockDim.x >> 5) + (threadIdx.x >> 5);
    const int nwave = gridDim.x * (blockDim.x >> 5);
    const long total = (long)Bn * Pn;

    float posC = 0.0f, posO = 0.0f;

    for (long a = wave; a < total; a += nwave) {
        const float* row = conf + (size_t)a * Cn;
        // prefetch the next row this wave will touch (3 cachelines)
        const float* nxt = conf + ((size_t)a + nwave) * Cn;
        if (lane < 3) __builtin_prefetch(nxt + lane * 32, 0, 1);

        float c0 = row[lane];
        float c1 = row[lane + 32];
        float c2 = (lane < 17) ? row[lane + 64] : -INFINITY;
        float m = fmaxf(fmaxf(c0, c1), c2);
        for (int off = 16; off; off >>= 1) m = fmaxf(m, __shfl_xor(m, off, 32));
        float s = expf(c0 - m) + expf(c1 - m) + ((lane < 17) ? expf(c2 - m) : 0.0f);
        for (int off = 16; off; off >>= 1) s += __shfl_xor(s, off, 32);

        if (lane == 0) {
            float o0 = obj[a * 2 + 0], o1 = obj[a * 2 + 1];
            float lse = m + logf(s);              // logsumexp(conf row)
            float col0 = o0 + lse;                // background logit
            float M = fmaxf(col0, o1 + m);
            float tot = expf(col0 - M) + s * expf(o1 + m - M);
            float lse_all = M + logf(tot);        // logsumexp over C+1 logits
            int t = conf_t[a];
            float logit_t = (t == 0) ? col0 : (o1 + row[t - 1]);
            float cec = lse_all - logit_t;

            float mo = fmaxf(o0, o1);
            float lseo = mo + logf(expf(o0 - mo) + expf(o1 - mo));
            bool pos = (t > 0);
            float ceo = lseo - (pos ? o1 : o0);

            masked_c[a] = pos ? 0.0f : cec;
            masked_o[a] = pos ? 0.0f : ceo;
            if (pos) { posC += cec; posO += ceo; }
        }
    }

    __shared__ float r0[256], r1[256];
    r0[threadIdx.x] = posC; r1[threadIdx.x] = posO;
    __syncthreads();
    for (int s2 = blockDim.x / 2; s2 > 0; s2 >>= 1) {
        if (threadIdx.x < s2) {
            r0[threadIdx.x] += r0[threadIdx.x + s2];
            r1[threadIdx.x] += r1[threadIdx.x + s2];
        }
        __syncthreads();
    }
    if (threadIdx.x == 0) {
        atomicAdd(&accum_f[1], r0[0]);
        atomicAdd(&accum_f[2], r1[0]);
    }
}

// ---------------------------------------------------------------------------
// K3: hard-negative mining via exact 3-level radix select on float bit
// patterns (losses >= 0 => IEEE bits are order-isomorphic). The mined
// quantity is a SUM, so tie-break-by-index is irrelevant:
//   topK_sum = sum(v : key > kthKey) + (K - count_greater) * kthValue.
// 4 passes over a 64 KB L2-resident row instead of O(P^2) compares.
// blockIdx.y selects conf (0) / obj (1) loss.
// ---------------------------------------------------------------------------
__global__ __launch_bounds__(256) void k_select(
    const float* __restrict__ masked_c, const float* __restrict__ masked_o,
    const int* __restrict__ num_pos, float* __restrict__ conf_neg,
    float* __restrict__ obj_neg) {
    const int b = blockIdx.x;
    const int tid = threadIdx.x;
    const float* row = (blockIdx.y == 0 ? masked_c : masked_o) + (size_t)b * Pn;
    float* outp = (blockIdx.y == 0 ? conf_neg : obj_neg);

    int K = num_pos[b] * 3;
    if (K > Pn - 1) K = Pn - 1;

    __shared__ int hist[2048];
    __shared__ int s_cgt;
    __shared__ unsigned s_prefix;
    __shared__ float rf[256];

    if (tid == 0) { s_cgt = 0; s_prefix = 0u; }

    // levels: bits [31:21] (2048 bins), [20:10] (2048), [9:0] (1024)
    for (int lvl = 0; lvl < 3; ++lvl) {
        for (int i = tid; i < 2048; i += 256) hist[i] = 0;
        __syncthreads();

        const unsigned pmask = (lvl == 0) ? 0u
                             : (lvl == 1) ? 0xFFE00000u : 0xFFFFFC00u;
        const int sh = (lvl == 0) ? 21 : (lvl == 1) ? 10 : 0;
        const int nb = (lvl == 2) ? 1024 : 2048;
        const unsigned pref = s_prefix;

        for (int p = tid; p < Pn; p += 256) {
            unsigned key = __float_as_uint(row[p]);
            if ((key & pmask) == pref)
                atomicAdd(&hist[(key >> sh) & (nb - 1)], 1);
        }
        __syncthreads();

        if (tid == 0) {
            int cum = s_cgt;
            int bsel = 0;
            for (int bin = nb - 1; bin >= 0; --bin) {
                if (cum + hist[bin] > K) { bsel = bin; break; }
                cum += hist[bin];
            }
            s_cgt = cum;                                  // strictly-greater count
            s_prefix = pref | ((unsigned)bsel << sh);     // refined key prefix
        }
        __syncthreads();
    }

    const unsigned thrKey = s_prefix;            // exact K-th largest key
    const int krem = K - s_cgt;                  // # boundary elements to take

    float local = 0.0f;
    for (int p = tid; p < Pn; p += 256) {
        float v = row[p];
        if (__float_as_uint(v) > thrKey) local += v;
    }
    rf[tid] = local;
    __syncthreads();
    for (int s = 128; s > 0; s >>= 1) {
        if (tid < s) rf[tid] += rf[tid + s];
        __syncthreads();
    }
    if (tid == 0) outp[b] = rf[0] + (float)krem * __uint_as_float(thrKey);
}

// ---------------------------------------------------------------------------
// K4: final reduction. The 64 per-batch negative sums are reduced with
// V_WMMA_F32_16X16X4_F32: A (16x4 f32, 2 regs/lane) = partial sums,
// B = ones => D rows are row-sums; sum D's 8 VGPRs + shfl_xor(16) = total.
// One full wave32, no divergence before WMMA (EXEC all-ones requirement).
// ---------------------------------------------------------------------------
__global__ __launch_bounds__(32) void k_final(
    const float* __restrict__ conf_neg, const float* __restrict__ obj_neg,
    const float* __restrict__ accum_f, const int* __restrict__ accum_i,
    float* __restrict__ out) {
    const int lane = threadIdx.x;
    v2f ones; ones.x = 1.0f; ones.y = 1.0f;
    v8f cz = {};

    v2f a;
    a.x = conf_neg[2 * lane]; a.y = conf_neg[2 * lane + 1];
    v8f d = __builtin_amdgcn_wmma_f32_16x16x4_f32(
        false, a, false, ones, (short)0, cz, false, false);
    float s = d[0] + d[1] + d[2] + d[3] + d[4] + d[5] + d[6] + d[7];
    float confTot = s + __shfl_xor(s, 16, 32);

    v2f a2;
    a2.x = obj_neg[2 * lane]; a2.y = obj_neg[2 * lane + 1];
    v8f d2 = __builtin_amdgcn_wmma_f32_16x16x4_f32(
        false, a2, false, ones, (short)0, cz, false, false);
    float s2 = d2[0] + d2[1] + d2[2] + d2[3] + d2[4] + d2[5] + d2[6] + d2[7];
    float objTot = s2 + __shfl_xor(s2, 16, 32);

    if (lane == 0) {
        int n = accum_i[0]; if (n < 1) n = 1;
        float N = (float)n;
        out[0] = accum_f[0] / N;                 // loss_l / N
        out[1] = (accum_f[1] + confTot) / N;     // loss_c / N
        out[2] = (accum_f[2] + objTot) / N;      // loss_obj / N
    }
}

// ---------------------------------------------------------------------------
extern "C" void kernel_launch(void* const* d_in, const int* in_sizes, int n_in,
                              void* d_out, int out_size, void* d_ws, size_t ws_size,
                              hipStream_t stream) {
    (void)in_sizes; (void)n_in; (void)out_size; (void)ws_size;

    const float* loc_data  = (const float*)d_in[0];   // [B,P,4]
    const float* conf_data = (const float*)d_in[1];   // [B,P,C]
    const float* obj_data  = (const float*)d_in[2];   // [B,P,2]
    const float* priors    = (const float*)d_in[3];   // [P,4]
    const float* targets   = (const float*)d_in[4];   // [B,O,5]
    float* out = (float*)d_out;

    const size_t SZ = (size_t)Bn * Pn * 4;            // bytes per B*P array
    char* ws = (char*)d_ws;
    int*   conf_t   = (int*)ws;                       // [B*P]
    float* masked_c = (float*)(ws + SZ);              // [B*P] (reused: tmp_ov)
    float* masked_o = (float*)(ws + 2 * SZ);          // [B*P] (reused: tmp_ti)
    int*   num_pos  = (int*)(ws + 3 * SZ);            // [B]
    float* accum_f  = (float*)(ws + 3 * SZ + 512);    // [8]
    int*   accum_i  = (int*)(ws + 3 * SZ + 768);      // [4]
    float* conf_neg = (float*)(ws + 3 * SZ + 1024);   // [B]
    float* obj_neg  = (float*)(ws + 3 * SZ + 1280);   // [B]

    k_zero<<<1, 128, 0, stream>>>(accum_f, accum_i, num_pos);

    k_match<<<Bn, 1024, 0, stream>>>(loc_data, priors, targets, conf_t,
                                     /*tmp_ov=*/masked_c, /*tmp_ti=*/(int*)masked_o,
                                     num_pos, accum_f, accum_i);

    k_ce<<<2048, 256, 0, stream>>>(conf_data, obj_data, conf_t,
                                   masked_c, masked_o, accum_f);

    k_select<<<dim3(Bn, 2), 256, 0, stream>>>(masked_c, masked_o, num_pos,
                                              conf_neg, obj_neg);

    k_final<<<1, 32, 0, stream>>>(conf_neg, obj_neg, accum_f, accum_i, out);
}